// GSNN_83330955477864
// MI455X (gfx1250) — compile-verified
//
#include <hip/hip_runtime.h>

typedef __attribute__((ext_vector_type(2))) float v2f;
typedef __attribute__((ext_vector_type(4))) float v4f;
typedef __attribute__((ext_vector_type(8))) float v8f;

constexpr int Bsz = 256;
constexpr int Nn  = 10000;
constexpr int Ee  = 40000;
constexpr int Cc  = 8;
constexpr int NLAYERS = 4;
constexpr float EPS = 1e-5f;

// ---------------- utility kernels ----------------

__global__ void k_zero_i32(int* __restrict__ p, int n) {
  int i = blockIdx.x * blockDim.x + threadIdx.x;
  if (i < n) p[i] = 0;
}

__global__ void k_zero_f32(float* __restrict__ p, int n) {
  int i = blockIdx.x * blockDim.x + threadIdx.x;
  if (i < n) p[i] = 0.f;
}

// in: R x C   ->  out: C x R
__global__ __launch_bounds__(256) void k_transpose(const float* __restrict__ in,
                                                   float* __restrict__ out,
                                                   int R, int C) {
  __shared__ float tile[32][33];
  int c0 = blockIdx.x * 32, r0 = blockIdx.y * 32;
  int tx = threadIdx.x;
  for (int i = threadIdx.y; i < 32; i += 8) {
    int r = r0 + i, c = c0 + tx;
    tile[i][tx] = (r < R && c < C) ? in[(size_t)r * C + c] : 0.f;
  }
  __syncthreads();
  int xo = r0 + tx;
  for (int i = threadIdx.y; i < 32; i += 8) {
    int co = c0 + i;
    if (co < C && xo < R) out[(size_t)co * R + xo] = tile[tx][i];
  }
}

// ---------------- CSR build (sort edges by dst) ----------------

__global__ void k_count(const int* __restrict__ dst, int* __restrict__ counts, int E) {
  int e = blockIdx.x * blockDim.x + threadIdx.x;
  if (e < E) atomicAdd(&counts[dst[e]], 1);
}

__global__ __launch_bounds__(256) void k_scan(const int* __restrict__ counts,
                                              int* __restrict__ offs, int n) {
  __shared__ int buf[256];
  __shared__ int carry;
  if (threadIdx.x == 0) { carry = 0; offs[0] = 0; }
  __syncthreads();
  for (int base = 0; base < n; base += 256) {
    int i = base + (int)threadIdx.x;
    int v = (i < n) ? counts[i] : 0;
    buf[threadIdx.x] = v;
    __syncthreads();
    for (int o = 1; o < 256; o <<= 1) {
      int t = (threadIdx.x >= (unsigned)o) ? buf[threadIdx.x - o] : 0;
      __syncthreads();
      buf[threadIdx.x] += t;
      __syncthreads();
    }
    int incl = buf[threadIdx.x];
    if (i < n) offs[i + 1] = carry + incl;
    __syncthreads();
    if (threadIdx.x == 255) carry += buf[255];
    __syncthreads();
  }
}

__global__ void k_fill(const int* __restrict__ dst, const int* __restrict__ offs,
                       int* __restrict__ cursor, int* __restrict__ eid, int E) {
  int e = blockIdx.x * blockDim.x + threadIdx.x;
  if (e < E) {
    int d = dst[e];
    int p = atomicAdd(&cursor[d], 1);
    eid[offs[d] + p] = e;
  }
}

// ---------------- edge init ----------------

__global__ __launch_bounds__(256) void k_init_edges(const float* __restrict__ xT,
                                                    const int* __restrict__ src,
                                                    float* __restrict__ x0,
                                                    float* __restrict__ xl) {
  int e = blockIdx.x, b = threadIdx.x;
  float v = xT[(size_t)src[e] * Bsz + b];
  size_t i = (size_t)e * Bsz + b;
  x0[i] = v;
  xl[i] = v;
}

// ---------------- BN stats: channel-splitting butterfly reduction ----------------
// Reduces sum & sumsq of v[0..7] over all 256 threads (batch axis).
// xor16/8/4 steps split channels across lanes while reducing rows (18 cross-lane
// ops per thread instead of 80), then tiny cross-wave combine in LDS.

__device__ __forceinline__ void bn_stats(const float* v, int lane, int wave, int tb,
                                         float red[2][8][8], float mus[8],
                                         float rss[8]) {
  float s8[8], q8[8];
#pragma unroll
  for (int c = 0; c < 8; c++) { s8[c] = v[c]; q8[c] = v[c] * v[c]; }

  float s4[4], q4[4];
  const bool k4 = (lane & 16) != 0;
#pragma unroll
  for (int j = 0; j < 4; j++) {
    float ms = k4 ? s8[4 + j] : s8[j];
    float os = k4 ? s8[j] : s8[4 + j];
    float mq = k4 ? q8[4 + j] : q8[j];
    float oq = k4 ? q8[j] : q8[4 + j];
    s4[j] = ms + __shfl_xor(os, 16, 32);
    q4[j] = mq + __shfl_xor(oq, 16, 32);
  }
  float s2[2], q2[2];
  const bool k3 = (lane & 8) != 0;
#pragma unroll
  for (int j = 0; j < 2; j++) {
    float ms = k3 ? s4[2 + j] : s4[j];
    float os = k3 ? s4[j] : s4[2 + j];
    float mq = k3 ? q4[2 + j] : q4[j];
    float oq = k3 ? q4[j] : q4[2 + j];
    s2[j] = ms + __shfl_xor(os, 8, 32);
    q2[j] = mq + __shfl_xor(oq, 8, 32);
  }
  const bool k2 = (lane & 4) != 0;
  float ms = k2 ? s2[1] : s2[0];
  float os = k2 ? s2[0] : s2[1];
  float mq = k2 ? q2[1] : q2[0];
  float oq = k2 ? q2[0] : q2[1];
  float s1 = ms + __shfl_xor(os, 4, 32);
  float q1 = mq + __shfl_xor(oq, 4, 32);
  // lane now owns channel c = lane[4:2], reduced over 8 rows; finish lane bits 0..1
  s1 += __shfl_xor(s1, 2, 32);
  q1 += __shfl_xor(q1, 2, 32);
  s1 += __shfl_xor(s1, 1, 32);
  q1 += __shfl_xor(q1, 1, 32);
  int c = (lane >> 2) & 7;
  if ((lane & 3) == 0) { red[0][wave][c] = s1; red[1][wave][c] = q1; }
  __syncthreads();
  if (tb < 8) {
    float ss = 0.f, qq = 0.f;
#pragma unroll
    for (int w = 0; w < 8; w++) { ss += red[0][w][tb]; qq += red[1][w][tb]; }
    float mu = ss * (1.f / (float)Bsz);
    float var = qq * (1.f / (float)Bsz) - mu * mu;
    mus[tb] = mu;
    rss[tb] = rsqrtf(var + EPS);
  }
  __syncthreads();
}

// ---------------- fused per-node layer kernel (steps 1-4) ----------------

__global__ __launch_bounds__(256) void k_layer_node(
    const float* __restrict__ xl, const int* __restrict__ offs,
    const int* __restrict__ eid, const float* __restrict__ W1v,
    const float* __restrict__ pb1, const float* __restrict__ pg1,
    const float* __restrict__ pbe1, const float* __restrict__ W2,
    const float* __restrict__ pb2, const float* __restrict__ pg2,
    const float* __restrict__ pbe2, const int* __restrict__ fnm,
    float* __restrict__ h) {
  const int n = blockIdx.x;
  const int tb = threadIdx.x;  // batch row
  const int lane = tb & 31, wave = tb >> 5;

  __shared__ float hs[256][8];    // BN1+ELU output (A-matrix source)
  __shared__ float h2s[256][8];   // einsum output
  __shared__ float w2raw[64];     // W2[n], filled by async DMA
  __shared__ float red[2][8][8];  // [sum|sumsq][wave][c]
  __shared__ float mus[8], rss[8];

  // ---- async-stage W2[n] (8x8 f32 = 256B) into LDS: 16 lanes x b128 ----
  if (tb < 16) {
    unsigned lds_off = (unsigned)(uintptr_t)(&w2raw[0]) + (unsigned)tb * 16u;
    unsigned long long ga =
        (unsigned long long)(uintptr_t)(W2 + (size_t)n * 64) + (unsigned long long)tb * 16ull;
    asm volatile("global_load_async_to_lds_b128 %0, %1, off"
                 :: "v"(lds_off), "v"(ga)
                 : "memory");
  }

  // ---- step 1: gather incoming edges (CSR) ----
  float acc[8];
#pragma unroll
  for (int c = 0; c < 8; c++) acc[c] = pb1[n * 8 + c];
  const int e0 = offs[n], e1 = offs[n + 1];
  for (int i = e0; i < e1; i++) {
    int e = eid[i];                       // wave-uniform
    float xv = xl[(size_t)e * Bsz + tb];  // coalesced
#pragma unroll
    for (int c = 0; c < 8; c++) acc[c] = fmaf(xv, W1v[e * 8 + c], acc[c]);
  }

  // ---- BN1 + ELU ----
  bn_stats(acc, lane, wave, tb, red, mus, rss);
#pragma unroll
  for (int c = 0; c < 8; c++) {
    float v = (acc[c] - mus[c]) * rss[c] * pg1[n * 8 + c] + pbe1[n * 8 + c];
    hs[tb][c] = (v > 0.f) ? v : expm1f(v);
  }

  // async W2 copy must be visible to all waves past this barrier
  asm volatile("s_wait_asynccnt 0x0" ::: "memory");
  __syncthreads();

  // ---- step 3: 256x8 @ 8x8 via V_WMMA_F32_16X16X4_F32 ----
  const float fnmv = fnm[n] ? 1.f : 0.f;
  const int hi = lane >> 4;   // K-half selector
  const int cl = lane & 15;   // M-row within tile (A) / N-col (B,D)
  const int ccl = cl & 7;     // clamped column for non-divergent loads
  const bool colok = (cl < 8);
  const float b2v = pb2[n * 8 + ccl];  // hoisted out of tile loop

  v2f bf0, bf1;
  bf0.x = colok ? w2raw[(0 + 2 * hi) * 8 + ccl] : 0.f;
  bf0.y = colok ? w2raw[(1 + 2 * hi) * 8 + ccl] : 0.f;
  bf1.x = colok ? w2raw[(4 + 2 * hi) * 8 + ccl] : 0.f;
  bf1.y = colok ? w2raw[(5 + 2 * hi) * 8 + ccl] : 0.f;

  for (int t = wave; t < 16; t += 8) {  // two 16-row tiles per wave, uniform
    int row = t * 16 + cl;
    v2f a0, a1;
    a0.x = hs[row][0 + 2 * hi];
    a0.y = hs[row][1 + 2 * hi];
    a1.x = hs[row][4 + 2 * hi];
    a1.y = hs[row][5 + 2 * hi];
    v8f d = {0.f, 0.f, 0.f, 0.f, 0.f, 0.f, 0.f, 0.f};
    d = __builtin_amdgcn_wmma_f32_16x16x4_f32(false, a0, false, bf0, (short)0, d,
                                              false, false);
    d = __builtin_amdgcn_wmma_f32_16x16x4_f32(false, a1, false, bf1, (short)0, d,
                                              false, false);
    if (colok) {
#pragma unroll
      for (int r = 0; r < 8; r++) {
        int m = t * 16 + r + 8 * hi;  // D layout: VGPR r -> M=r (+8 for hi lanes)
        h2s[m][cl] = d[r] * fnmv + b2v;
      }
    }
  }
  __syncthreads();

  // ---- BN2 + ELU + store h ----
  float a2[8];
#pragma unroll
  for (int c = 0; c < 8; c++) a2[c] = h2s[tb][c];
  bn_stats(a2, lane, wave, tb, red, mus, rss);
  v4f o0, o1;
#pragma unroll
  for (int c = 0; c < 8; c++) {
    float v = (a2[c] - mus[c]) * rss[c] * pg2[n * 8 + c] + pbe2[n * 8 + c];
    v = (v > 0.f) ? v : expm1f(v);
    if (c < 4) o0[c] = v; else o1[c - 4] = v;
  }
  v4f* hp = (v4f*)(h + ((size_t)n * Bsz + tb) * 8);
  hp[0] = o0;
  hp[1] = o1;
}

// ---------------- step 5: per-edge gather dot + residual/EMA ----------------

__global__ __launch_bounds__(256) void k_edge_update(
    const float* __restrict__ h, const int* __restrict__ src,
    const float* __restrict__ W3v, const float* __restrict__ b3,
    const int* __restrict__ fnm, const float* __restrict__ x0,
    float* __restrict__ xl, const float* __restrict__ alpha) {
  int e = blockIdx.x, b = threadIdx.x;
  int s = src[e];
  const v4f* hp = (const v4f*)(h + ((size_t)s * Bsz + b) * 8);
  v4f h0 = hp[0], h1 = hp[1];
  float dot = 0.f;
#pragma unroll
  for (int c = 0; c < 4; c++) dot = fmaf(h0[c], W3v[e * 8 + c], dot);
#pragma unroll
  for (int c = 0; c < 4; c++) dot = fmaf(h1[c], W3v[e * 8 + 4 + c], dot);
  float fm = fnm[s] ? 1.f : 0.f;
  float ev = dot * fm + b3[e];
  float a = 1.f / (1.f + __expf(-alpha[0]));
  size_t idx = (size_t)e * Bsz + b;
  float v = (1.f - a) * (ev + x0[idx]) + a * xl[idx];
  xl[idx] = v;
}

// ---------------- final output scatter (set semantics) ----------------

__global__ __launch_bounds__(256) void k_scatter_out(const int* __restrict__ dst,
                                                     const int* __restrict__ omask,
                                                     const float* __restrict__ xl,
                                                     float* __restrict__ outT) {
  int e = blockIdx.x, b = threadIdx.x;
  int d = dst[e];
  if (omask[d]) outT[(size_t)d * Bsz + b] = xl[(size_t)e * Bsz + b];
}

// ---------------- host launcher ----------------

extern "C" void kernel_launch(void* const* d_in, const int* in_sizes, int n_in,
                              void* d_out, int out_size, void* d_ws, size_t ws_size,
                              hipStream_t stream) {
  (void)in_sizes; (void)n_in; (void)out_size; (void)ws_size;
  const float* x   = (const float*)d_in[0];
  const int* ei    = (const int*)d_in[1];
  const int* src   = ei;
  const int* dst   = ei + Ee;
  const int* fnm   = (const int*)d_in[2];
  const int* onm   = (const int*)d_in[3];
  const float* W1v = (const float*)d_in[4];
  const float* b1  = (const float*)d_in[5];
  const float* g1  = (const float*)d_in[6];
  const float* be1 = (const float*)d_in[7];
  const float* W2  = (const float*)d_in[8];
  const float* b2  = (const float*)d_in[9];
  const float* g2  = (const float*)d_in[10];
  const float* be2 = (const float*)d_in[11];
  const float* W3v = (const float*)d_in[12];
  const float* b3  = (const float*)d_in[13];
  const float* alpha = (const float*)d_in[14];
  float* out = (float*)d_out;

  char* ws = (char*)d_ws;
  size_t off = 0;
  auto take = [&](size_t bytes) -> void* {
    void* p = ws + off;
    off += (bytes + 255) & ~(size_t)255;
    return p;
  };
  float* xT   = (float*)take(sizeof(float) * (size_t)Nn * Bsz);
  float* x0   = (float*)take(sizeof(float) * (size_t)Ee * Bsz);
  float* xl   = (float*)take(sizeof(float) * (size_t)Ee * Bsz);
  float* h    = (float*)take(sizeof(float) * (size_t)Nn * Bsz * Cc);
  int* counts = (int*)take(sizeof(int) * Nn);
  int* cursor = (int*)take(sizeof(int) * Nn);
  int* offs   = (int*)take(sizeof(int) * (Nn + 1));
  int* eid    = (int*)take(sizeof(int) * Ee);
  float* outT = xT;  // xT dead after edge init; reuse for output staging

  // CSR build
  k_zero_i32<<<(Nn + 255) / 256, 256, 0, stream>>>(counts, Nn);
  k_zero_i32<<<(Nn + 255) / 256, 256, 0, stream>>>(cursor, Nn);
  k_count<<<(Ee + 255) / 256, 256, 0, stream>>>(dst, counts, Ee);
  k_scan<<<1, 256, 0, stream>>>(counts, offs, Nn);
  k_fill<<<(Ee + 255) / 256, 256, 0, stream>>>(dst, offs, cursor, eid, Ee);

  // x (B x N) -> xT (N x B); edge init
  {
    dim3 blk(32, 8);
    dim3 grd((Nn + 31) / 32, (Bsz + 31) / 32);
    k_transpose<<<grd, blk, 0, stream>>>(x, xT, Bsz, Nn);
  }
  k_init_edges<<<Ee, 256, 0, stream>>>(xT, src, x0, xl);

  for (int l = 0; l < NLAYERS; l++) {
    k_layer_node<<<Nn, 256, 0, stream>>>(xl, offs, eid, W1v, b1, g1, be1, W2, b2,
                                         g2, be2, fnm, h);
    k_edge_update<<<Ee, 256, 0, stream>>>(h, src, W3v, b3, fnm, x0, xl, alpha);
  }

  k_zero_f32<<<((Nn * Bsz) + 255) / 256, 256, 0, stream>>>(outT, Nn * Bsz);
  k_scatter_out<<<Ee, 256, 0, stream>>>(dst, onm, xl, outT);
  {
    dim3 blk(32, 8);
    dim3 grd((Bsz + 31) / 32, (Nn + 31) / 32);
    k_transpose<<<grd, blk, 0, stream>>>(outT, out, Nn, Bsz);
  }
}